// RecurrentInhibition_9981503996084
// MI455X (gfx1250) — compile-verified
//
#include <hip/hip_runtime.h>

// ---------------------------------------------------------------------------
// RecurrentInhibition on gfx1250 (MI455X)
//   63-tap conv along channels as fp32 WMMA (V_WMMA_F32_16X16X4_F32).
//   Each wave: 4 accumulators (64 output channels) x 16 columns, sharing one
//   128-channel halo window -> 80 WMMAs per 64 B-loads. Interior waves use
//   immediate-offset loads; boundary waves clamp+cndmask. launch_bounds caps
//   VGPRs at 256/wave (>=4 waves/SIMD). Deterministic 2-stage MSE reduction
//   drives a device-side sticky convergence flag (graph-capture safe).
// ---------------------------------------------------------------------------

typedef __attribute__((ext_vector_type(2))) float v2f;
typedef __attribute__((ext_vector_type(8))) float v8f;

#define NB      16
#define CCH     512
#define HW      1024                     // 32*32
#define TOTAL   (NB * CCH * HW)          // 8388608
#define SCOPE   63
#define PAD     31
#define NCH     32                       // K=4 chunks per 128-ch halo window
#define NTA     20                       // banded A matrices per 16-out tile
#define NSTEPS  20
#define CONV_TH 0.001f
#define NBLOCKS 1024

// ws layout
#define WS_DONE_OFF   0
#define WS_PART_OFF   64                       // 1024 block partial sums
#define WS_INHIB_OFF  8192
#define WS_STATE_OFF  (8192 + (size_t)TOTAL * 4)

__global__ void ri_init_kernel(int* done) {
    if (threadIdx.x == 0) *done = 0;
}

// mode 0: out = conv(x, w)                         (initial inhibition)
// mode 1: cand = 0.95*state + 0.025*(inhib+conv);  gated by *done; MSE partial
__global__ __launch_bounds__(256, 4) void ri_conv_step_kernel(
    const float* __restrict__ x,        // state (or activations, mode 0)
    const float* __restrict__ w,        // 63 taps
    const float* __restrict__ inhib,    // mode 1 only
    float* __restrict__ out,
    const int* __restrict__ doneFlag,   // mode 1 only
    float* __restrict__ partial,        // mode 1 only: [NBLOCKS]
    int mode)
{
    const int lane   = threadIdx.x & 31;
    const int i      = lane & 15;            // A row index (M) / column (N)
    const int kb     = (lane >> 4) << 1;     // K sub-index base: 0 or 2
    const int rowOff = (lane >> 4) << 3;     // C layout: +8 rows for hi half

    const int d0    = (threadIdx.x >> 5) << 6;       // 0,64,...,448
    const int batch = blockIdx.x >> 6;               // 64 hw-tiles per batch
    const int hw0   = (blockIdx.x & 63) << 4;
    const size_t colBase = (size_t)batch * (CCH * HW) + hw0 + i;

    const bool isDone = (mode != 0) && (*doneFlag != 0);
    float localSum = 0.0f;

    if (isDone) {   // frozen: propagate state through ping-pong chain
#pragma unroll
        for (int r = 0; r < 4; ++r)
#pragma unroll
            for (int row = 0; row < 8; ++row) {
                size_t off = colBase + (size_t)(d0 + 16*r + row + rowOff) * HW;
                out[off] = x[off];
            }
    } else {
        // --- banded A matrices: A_t[i][k] = w[4t+k-i] for idx in [0,63) ---
        v2f A[NTA];
#pragma unroll
        for (int t = 0; t < NTA; ++t) {
            int j0 = 4 * t + kb - i;
            int j1 = j0 + 1;
            A[t][0] = (j0 >= 0 && j0 < SCOPE) ? w[j0] : 0.0f;
            A[t][1] = (j1 >= 0 && j1 < SCOPE) ? w[j1] : 0.0f;
        }

        v8f acc[4] = {v8f{}, v8f{}, v8f{}, v8f{}};

        if (d0 >= PAD && d0 + 96 < CCH) {
            // ---- interior fast path: immediate-offset loads only ----
            const float* p = x + colBase + (size_t)(d0 - PAD + kb) * HW;
#pragma unroll
            for (int t = 0; t < NCH; ++t) {
                v2f B;
                B[0] = p[(4 * t)     * HW];
                B[1] = p[(4 * t + 1) * HW];
#pragma unroll
                for (int r = 0; r < 4; ++r)
                    if (t >= 4 * r && t < 4 * r + NTA)
                        acc[r] = __builtin_amdgcn_wmma_f32_16x16x4_f32(
                            false, A[t - 4 * r], false, B, (short)0, acc[r],
                            false, false);
            }
        } else {
            // ---- boundary path: clamp + cndmask zero padding ----
#pragma unroll
            for (int t = 0; t < NCH; ++t) {
                int s0 = d0 - PAD + 4 * t + kb;
                int s1 = s0 + 1;
                int c0 = min(max(s0, 0), CCH - 1);
                int c1 = min(max(s1, 0), CCH - 1);
                float b0 = x[colBase + (size_t)c0 * HW];
                float b1 = x[colBase + (size_t)c1 * HW];
                v2f B;
                B[0] = (s0 == c0) ? b0 : 0.0f;
                B[1] = (s1 == c1) ? b1 : 0.0f;
#pragma unroll
                for (int r = 0; r < 4; ++r)
                    if (t >= 4 * r && t < 4 * r + NTA)
                        acc[r] = __builtin_amdgcn_wmma_f32_16x16x4_f32(
                            false, A[t - 4 * r], false, B, (short)0, acc[r],
                            false, false);
            }
        }

        if (mode == 0) {
#pragma unroll
            for (int r = 0; r < 4; ++r)
#pragma unroll
                for (int row = 0; row < 8; ++row) {
                    size_t off =
                        colBase + (size_t)(d0 + 16*r + row + rowOff) * HW;
                    out[off] = acc[r][row];
                }
        } else {
#pragma unroll
            for (int r = 0; r < 4; ++r)
#pragma unroll
                for (int row = 0; row < 8; ++row) {
                    size_t off =
                        colBase + (size_t)(d0 + 16*r + row + rowOff) * HW;
                    float st   = x[off];
                    float phi  = 0.5f * (inhib[off] + acc[r][row]);
                    float cand = 0.95f * st + 0.05f * phi;
                    float d    = cand - st;
                    localSum  += d * d;
                    out[off]   = cand;
                }
        }
    }

    if (mode != 0) {   // deterministic block tree-reduce, no atomics
        __shared__ float red[256];
        red[threadIdx.x] = localSum;
        __syncthreads();
        for (int s = 128; s > 0; s >>= 1) {
            if (threadIdx.x < s) red[threadIdx.x] += red[threadIdx.x + s];
            __syncthreads();
        }
        if (threadIdx.x == 0) partial[blockIdx.x] = red[0];
    }
}

// Deterministic final reduction of block partials -> sticky done flag.
__global__ void ri_control_kernel(const float* __restrict__ partial,
                                  int* __restrict__ done) {
    __shared__ float red[256];
    float s = 0.0f;
    for (int k = threadIdx.x; k < NBLOCKS; k += 256) s += partial[k];
    red[threadIdx.x] = s;
    __syncthreads();
    for (int t = 128; t > 0; t >>= 1) {
        if (threadIdx.x < t) red[threadIdx.x] += red[threadIdx.x + t];
        __syncthreads();
    }
    if (threadIdx.x == 0) {
        float diff = red[0] * (1.0f / (float)TOTAL);
        if (diff <= CONV_TH) *done = 1;   // sticky
    }
}

extern "C" void kernel_launch(void* const* d_in, const int* in_sizes, int n_in,
                              void* d_out, int out_size, void* d_ws, size_t ws_size,
                              hipStream_t stream) {
    (void)in_sizes; (void)n_in; (void)out_size; (void)ws_size;

    const float* acts  = (const float*)d_in[0];
    const float* w_in  = (const float*)d_in[1];
    const float* w_rec = (const float*)d_in[2];
    float*       out   = (float*)d_out;
    char*        ws    = (char*)d_ws;

    int*   done    = (int*)(ws + WS_DONE_OFF);
    float* partial = (float*)(ws + WS_PART_OFF);
    float* inhib   = (float*)(ws + WS_INHIB_OFF);
    float* stateB  = (float*)(ws + WS_STATE_OFF);

    ri_init_kernel<<<1, 32, 0, stream>>>(done);

    const dim3 grid(NBLOCKS), block(256); // block = 512-ch stripe x 16 cols

    // inhib_in = conv(activations, w_in); also serves as state s0
    ri_conv_step_kernel<<<grid, block, 0, stream>>>(
        acts, w_in, nullptr, inhib, nullptr, nullptr, 0);

    const float* cur = inhib;
    for (int s = 0; s < NSTEPS; ++s) {
        float* nxt = (s & 1) ? out : stateB;   // step 19 (odd) lands in d_out
        ri_conv_step_kernel<<<grid, block, 0, stream>>>(
            cur, w_rec, inhib, nxt, done, partial, 1);
        ri_control_kernel<<<1, 256, 0, stream>>>(partial, done);
        cur = nxt;
    }
}